// QuantumClassifierBackend_12120397710209
// MI455X (gfx1250) — compile-verified
//
#include <hip/hip_runtime.h>
#include <math.h>

// ---------------------------------------------------------------------------
// Quantum classifier: 10 qubits, 3 layers, batch 2048.
// One wave32 per sample. State (1024 cplx) in LDS. Fused RX*Rot gates,
// CNOT ring, PauliZ logits, rho = psi psi^H via V_WMMA_F32_16X16X4_F32,
// 32x32 complex Hermitian Jacobi eigensolve (trig-free rotations),
// von Neumann entropy. Transcendentals use native v_sin/v_cos/v_exp/v_log.
// ---------------------------------------------------------------------------

typedef float v2f __attribute__((ext_vector_type(2)));
typedef float v8f __attribute__((ext_vector_type(8)));

#define NQ 10
#define NL 3
#define NC 3
#define NSTATE 1024
#define JACOBI_SWEEPS 8

__device__ __forceinline__ void lds_fence() {
    // Same-wave DS ops are in-order on CDNA5; this is a compiler ordering
    // fence + explicit dscnt drain so cross-lane LDS RAW is safe intra-wave.
    asm volatile("s_wait_dscnt 0" ::: "memory");
}

__device__ __forceinline__ float2 cxmul(float2 a, float2 b) {
    return make_float2(a.x * b.x - a.y * b.y, a.x * b.y + a.y * b.x);
}
__device__ __forceinline__ float2 cxadd(float2 a, float2 b) {
    return make_float2(a.x + b.x, a.y + b.y);
}
// z * (-i*rs)
__device__ __forceinline__ float2 cx_mis(float2 z, float rs) {
    return make_float2(z.y * rs, -z.x * rs);
}

// fast tanh via native exp: tanh(x) = (e^{2x}-1)/(e^{2x}+1), clamped
__device__ __forceinline__ float fast_tanhf(float x) {
    float xc = fminf(fmaxf(x, -10.0f), 10.0f);
    float e  = __expf(2.0f * xc);
    return (e - 1.0f) / (e + 1.0f);
}

__device__ __forceinline__ float wave_sum32(float v) {
    #pragma unroll
    for (int m = 16; m > 0; m >>= 1) v += __shfl_xor(v, m, 32);
    return v;
}

__device__ __forceinline__ v8f wmma_f32(v2f a, v2f b, v8f c) {
    return __builtin_amdgcn_wmma_f32_16x16x4_f32(
        /*neg_a=*/false, a, /*neg_b=*/false, b,
        /*c_mod=*/(short)0, c, /*reuse_a=*/false, /*reuse_b=*/false);
}

__global__ __launch_bounds__(32)
void QuantumClassifierBackend_12120397710209_kernel(
        const float* __restrict__ X,    // (B, 10)
        const float* __restrict__ W,    // (3, 10, 3)
        float* __restrict__ OUT,        // logits (B,3) flat, then entropy (B)
        int B) {
    __shared__ __align__(16) float2 S[NSTATE];   // state, later rho (32x32)

    const int lane = threadIdx.x;   // 0..31 (wave32)
    const int b    = blockIdx.x;    // sample index
    const float PI = 3.14159265358979323846f;

    // ---- init |0...0> ----
    #pragma unroll 4
    for (int q = 0; q < 32; ++q) {
        int i = q * 32 + lane;
        S[i] = (i == 0) ? make_float2(1.0f, 0.0f) : make_float2(0.0f, 0.0f);
    }
    lds_fence();

    // ---- per-wire RX coefficients (layer-invariant): cos/sin(tanh(x)*pi/2)
    float rcArr[NQ], rsArr[NQ];
    #pragma unroll
    for (int w = 0; w < NQ; ++w) {
        float half = 0.5f * (fast_tanhf(X[b * NQ + w]) * PI);
        __sincosf(half, &rsArr[w], &rcArr[w]);
    }

    // ---- circuit ----
    for (int layer = 0; layer < NL; ++layer) {
        // fused G = Rot(phi,theta,omega) * RX per wire
        for (int w = 0; w < NQ; ++w) {
            float rc = rcArr[w], rs = rsArr[w];

            const float* wp = W + (layer * NQ + w) * 3;
            float phi = wp[0], th = wp[1], om = wp[2];
            float ct, st;
            __sincosf(0.5f * th, &st, &ct);
            float al = 0.5f * (phi + om), be = 0.5f * (phi - om);
            float sa, ca, sb, cb;
            __sincosf(al, &sa, &ca);
            __sincosf(be, &sb, &cb);
            float2 ep = make_float2(ca, -sa);   // e^{-i(phi+om)/2}
            float2 em = make_float2(cb, -sb);   // e^{-i(phi-om)/2}
            float2 r00 = make_float2( ct * ep.x,  ct * ep.y);
            float2 r01 = make_float2(-st * em.x,  st * em.y);  // -conj(em)*st
            float2 r10 = make_float2( st * em.x,  st * em.y);
            float2 r11 = make_float2( ct * ep.x, -ct * ep.y);  // conj(ep)*ct

            // G = R * RX,  RX = [[rc, -i*rs],[-i*rs, rc]]
            float2 g00 = cxadd(make_float2(rc * r00.x, rc * r00.y), cx_mis(r01, rs));
            float2 g01 = cxadd(cx_mis(r00, rs), make_float2(rc * r01.x, rc * r01.y));
            float2 g10 = cxadd(make_float2(rc * r10.x, rc * r10.y), cx_mis(r11, rs));
            float2 g11 = cxadd(cx_mis(r10, rs), make_float2(rc * r11.x, rc * r11.y));

            int stride = 1 << (9 - w);
            #pragma unroll 4
            for (int p = 0; p < 16; ++p) {
                int pi  = p * 32 + lane;            // pair index 0..511
                int low = pi & (stride - 1);
                int i0  = ((pi ^ low) << 1) | low;  // bit (9-w) == 0
                int i1  = i0 | stride;
                float2 a0 = S[i0];
                float2 a1 = S[i1];
                float2 n0 = cxadd(cxmul(g00, a0), cxmul(g01, a1));
                float2 n1 = cxadd(cxmul(g10, a0), cxmul(g11, a1));
                S[i0] = n0;
                S[i1] = n1;
            }
            lds_fence();
        }
        // CNOT ring: control w, target (w+1)%10
        for (int w = 0; w < NQ; ++w) {
            int c = w, t = (w + 1) % NQ;
            int bc = 9 - c, bt = 9 - t;
            int hi = bc > bt ? bc : bt;
            int lo = bc < bt ? bc : bt;
            int mid = hi - lo - 1;
            #pragma unroll 4
            for (int q = 0; q < 8; ++q) {
                int s8   = q * 32 + lane;                 // 0..255
                int m1   = s8 & ((1 << lo) - 1);
                int rest = s8 >> lo;
                int m2   = rest & ((1 << mid) - 1);
                int m3   = rest >> mid;
                int base = (m3 << (hi + 1)) | (m2 << (lo + 1)) | m1;
                int iA   = base | (1 << bc);              // control=1, target=0
                int iB   = iA | (1 << bt);                // target=1
                float2 va = S[iA];
                float2 vb = S[iB];
                S[iA] = vb;
                S[iB] = va;
            }
            lds_fence();
        }
    }

    // ---- logits: <Z_w> for w=0..2 (bit positions 9,8,7) ----
    float l0 = 0.f, l1 = 0.f, l2 = 0.f;
    #pragma unroll 4
    for (int q = 0; q < 32; ++q) {
        int a = q * 32 + lane;
        float2 v = S[a];
        float pr = v.x * v.x + v.y * v.y;
        l0 += ((a >> 9) & 1) ? -pr : pr;
        l1 += ((a >> 8) & 1) ? -pr : pr;
        l2 += ((a >> 7) & 1) ? -pr : pr;
    }
    l0 = wave_sum32(l0); l1 = wave_sum32(l1); l2 = wave_sum32(l2);
    if (lane == 0) {
        OUT[b * NC + 0] = l0;
        OUT[b * NC + 1] = l1;
        OUT[b * NC + 2] = l2;
    }

    // ---- rho = psi psi^H  (psi: 32x32 complex, row-major in S) ----
    // rho_re = Pr Pr^T + Pi Pi^T ; rho_im = Pi Pr^T - Pr Pi^T
    // WMMA f32 16x16x4: A(m,k)/B(k,n) frag at lane = idx%16 + 16*(k_local>>1),
    // vgpr = k_local&1. One ds_load_b128 yields {re0,im0,re1,im1}.
    v8f re00 = {0,0,0,0,0,0,0,0}, re01 = {0,0,0,0,0,0,0,0};
    v8f re10 = {0,0,0,0,0,0,0,0}, re11 = {0,0,0,0,0,0,0,0};
    v8f im00 = {0,0,0,0,0,0,0,0}, im01 = {0,0,0,0,0,0,0,0};
    v8f im10 = {0,0,0,0,0,0,0,0}, im11 = {0,0,0,0,0,0,0,0};

    const int r0  = lane & 15;
    const int kb2 = (lane >> 4) << 1;   // 0 or 2

    for (int k0 = 0; k0 < 32; k0 += 4) {
        int kb = k0 + kb2;
        float4 f0 = *reinterpret_cast<const float4*>(&S[r0 * 32 + kb]);
        float4 f1 = *reinterpret_cast<const float4*>(&S[(16 + r0) * 32 + kb]);
        v2f a0r = {f0.x, f0.z}, a0i = {f0.y, f0.w};
        v2f a1r = {f1.x, f1.z}, a1i = {f1.y, f1.w};
        v2f n0r = {-f0.x, -f0.z};
        v2f n1r = {-f1.x, -f1.z};

        // tile (0,0)
        re00 = wmma_f32(a0r, a0r, re00); re00 = wmma_f32(a0i, a0i, re00);
        im00 = wmma_f32(a0i, a0r, im00); im00 = wmma_f32(n0r, a0i, im00);
        // tile (0,16)
        re01 = wmma_f32(a0r, a1r, re01); re01 = wmma_f32(a0i, a1i, re01);
        im01 = wmma_f32(a0i, a1r, im01); im01 = wmma_f32(n0r, a1i, im01);
        // tile (16,0)
        re10 = wmma_f32(a1r, a0r, re10); re10 = wmma_f32(a1i, a0i, re10);
        im10 = wmma_f32(a1i, a0r, im10); im10 = wmma_f32(n1r, a0i, im10);
        // tile (16,16)
        re11 = wmma_f32(a1r, a1r, re11); re11 = wmma_f32(a1i, a1i, re11);
        im11 = wmma_f32(a1i, a1r, im11); im11 = wmma_f32(n1r, a1i, im11);
    }
    lds_fence();

    // store rho over S (D layout: row = v + 8*(lane/16), col = lane%16)
    {
        int cx = lane & 15;
        int rh = (lane >> 4) * 8;
        #pragma unroll
        for (int v = 0; v < 8; ++v) {
            int rr = v + rh;
            S[rr * 32 + cx]             = make_float2(re00[v], im00[v]);
            S[rr * 32 + 16 + cx]        = make_float2(re01[v], im01[v]);
            S[(16 + rr) * 32 + cx]      = make_float2(re10[v], im10[v]);
            S[(16 + rr) * 32 + 16 + cx] = make_float2(re11[v], im11[v]);
        }
    }
    lds_fence();

    // ---- cyclic complex Jacobi on 32x32 Hermitian rho in S (trig-free) ----
    for (int sweep = 0; sweep < JACOBI_SWEEPS; ++sweep) {
        for (int p = 0; p < 31; ++p) {
            for (int q = p + 1; q < 32; ++q) {
                float2 apq = S[p * 32 + q];           // uniform broadcast loads
                float  app = S[p * 32 + p].x;
                float  aqq = S[q * 32 + q].x;
                float  r2  = apq.x * apq.x + apq.y * apq.y;
                if (r2 > 1e-24f) {                    // wave-uniform branch
                    float r   = sqrtf(r2);
                    // algebraic small-angle Jacobi rotation:
                    // tau = cot(2*theta), t = tan(theta), |theta| <= pi/4
                    float tau = (aqq - app) / (2.0f * r);
                    float t   = copysignf(1.0f, tau) /
                                (fabsf(tau) + sqrtf(1.0f + tau * tau));
                    float c   = 1.0f / sqrtf(1.0f + t * t);
                    float sg  = t * c;
                    float si  = sg / r;
                    float2 s  = make_float2(si * apq.x, si * apq.y); // sg*e^{i phi}

                    // column update: A <- A*J   (lane = row i)
                    float2 vp = S[lane * 32 + p];
                    float2 vq = S[lane * 32 + q];
                    float2 np = make_float2(c * vp.x - (s.x * vq.x + s.y * vq.y),
                                            c * vp.y - (s.x * vq.y - s.y * vq.x));
                    float2 nq = make_float2(c * vq.x + (s.x * vp.x - s.y * vp.y),
                                            c * vq.y + (s.x * vp.y + s.y * vp.x));
                    S[lane * 32 + p] = np;
                    S[lane * 32 + q] = nq;
                    lds_fence();

                    // row update: A <- J^H * A  (lane = col j)
                    float2 wp = S[p * 32 + lane];
                    float2 wq = S[q * 32 + lane];
                    float2 mp = make_float2(c * wp.x - (s.x * wq.x - s.y * wq.y),
                                            c * wp.y - (s.x * wq.y + s.y * wq.x));
                    float2 mq = make_float2(c * wq.x + (s.x * wp.x + s.y * wp.y),
                                            c * wq.y + (s.x * wp.y - s.y * wp.x));
                    S[p * 32 + lane] = mp;
                    S[q * 32 + lane] = mq;
                    lds_fence();
                }
            }
        }
    }

    // ---- entropy from eigenvalues (diagonal) ----
    float ev   = S[lane * 32 + lane].x;
    float term = (ev > 1e-8f) ? (-ev * __logf(ev)) : 0.0f;
    float ent  = wave_sum32(term);
    if (lane == 0) OUT[B * NC + b] = ent;
}

extern "C" void kernel_launch(void* const* d_in, const int* in_sizes, int n_in,
                              void* d_out, int out_size, void* d_ws, size_t ws_size,
                              hipStream_t stream) {
    const float* X = (const float*)d_in[0];   // (B,10) f32
    const float* W = (const float*)d_in[1];   // (3,10,3) f32
    float* OUT = (float*)d_out;               // (B*3 logits) ++ (B entropy)
    int B = in_sizes[0] / NQ;
    QuantumClassifierBackend_12120397710209_kernel<<<B, 32, 0, stream>>>(X, W, OUT, B);
}